// GraphBuilder_20916490731746
// MI455X (gfx1250) — compile-verified
//
#include <hip/hip_runtime.h>
#include <hip/hip_bf16.h>
#include <math.h>

// ---------------- problem constants ----------------
#define NB       4096          // current batch rows
#define NH       4096          // history rows (128*32)
#define NN       8192          // total nodes
#define CF       256           // feature dim
#define CP       32            // pos-enc dim (4 coords * 2 * 4 freqs)
#define CG       288           // fused dim
#define TOPK     16
#define EPSV     1e-12f
#define SQRT_A   0.9746794f    // sqrt(0.95)
#define SQRT_1MA 0.22360680f   // sqrt(0.05)
#define TWO_PI   6.283185307179586f
#define SENTINEL (-3.4e38f)

typedef __attribute__((ext_vector_type(2))) float v2f;
typedef __attribute__((ext_vector_type(4))) float f4;
typedef __attribute__((ext_vector_type(8))) float v8f;

// ---------------------------------------------------------------------------
// K1: build fused normalized matrix G[8192][288] and copy node_feats output.
// One block (256 threads) per row.
// ---------------------------------------------------------------------------
__global__ __launch_bounds__(256) void prep_kernel(
    const float* __restrict__ feats, const float* __restrict__ coords,
    const float* __restrict__ hfeats, const float* __restrict__ hcoords,
    float* __restrict__ G, float* __restrict__ node_out)
{
    const int row = blockIdx.x;
    const int t   = threadIdx.x;

    const float* f = (row < NB) ? (feats  + (size_t)row * CF)
                                : (hfeats + (size_t)(row - NB) * CF);
    const float* c = (row < NB) ? (coords  + (size_t)row * 4)
                                : (hcoords + (size_t)(row - NB) * 4);

    float v = f[t];
    node_out[(size_t)row * CF + t] = v;           // node_feats output (raw)

    __shared__ float red[256];
    __shared__ float pe[CP];
    __shared__ float snorm[2];

    red[t] = v * v;
    __syncthreads();
    for (int s = 128; s > 0; s >>= 1) {
        if (t < s) red[t] += red[t + s];
        __syncthreads();
    }
    if (t == 0) snorm[0] = fmaxf(sqrtf(red[0]), EPSV);

    if (t < CP) {                                  // pos enc, one wave
        int d = t >> 3, r = t & 7, fr = r & 3;
        float ang = c[d] * TWO_PI * (float)(1 << fr);
        pe[t] = (r < 4) ? sinf(ang) : cosf(ang);
    }
    __syncthreads();
    if (t == 0) {
        float s2 = 0.f;
        for (int i = 0; i < CP; ++i) s2 += pe[i] * pe[i];
        snorm[1] = fmaxf(sqrtf(s2), EPSV);
    }
    __syncthreads();

    float* g = G + (size_t)row * CG;
    g[t] = v * (SQRT_A / snorm[0]);
    if (t < CP) g[CF + t] = pe[t] * (SQRT_1MA / snorm[1]);
}

// ---------------------------------------------------------------------------
// K2: S = G * G^T via V_WMMA_F32_16X16X4_F32.
// Workgroup: 256 threads = 8 waves, computes 128x128 output tile.
// Each wave: 32 rows x 64 cols = 8 accumulators (2 m-tiles x 4 n-tiles).
// LDS chunks 128x32 with stride-36 float padding (conflict-free).
// Cross-block (cur vs hist) uniform 0.5 scale applied at store.
// ---------------------------------------------------------------------------
#define LST 36

__global__ __launch_bounds__(256) void gemm_kernel(
    const float* __restrict__ G, float* __restrict__ S)
{
    __shared__ float lA[128 * LST];
    __shared__ float lB[128 * LST];

    const int t    = threadIdx.x;
    const int bx   = blockIdx.x, by = blockIdx.y;
    const int wave = t >> 5, lane = t & 31;
    const int wr   = wave >> 1, wc = wave & 1;       // 4x2 wave grid
    const int l15  = lane & 15, kh = lane >> 4;

    const int rowBase = by * 128;
    const int colBase = bx * 128;

    v8f acc[2][4];
    for (int i = 0; i < 2; ++i)
        for (int j = 0; j < 4; ++j)
            acc[i][j] = (v8f){0.f,0.f,0.f,0.f,0.f,0.f,0.f,0.f};

    const int rr   = t >> 1;        // 0..127: row within chunk
    const int half = t & 1;         // which 16-float half of the 32-wide chunk

    for (int kc = 0; kc < CG; kc += 32) {
        const float* pa = G + (size_t)(rowBase + rr) * CG + kc + half * 16;
        const float* pb = G + (size_t)(colBase + rr) * CG + kc + half * 16;
        float* la = lA + rr * LST + half * 16;
        float* lb = lB + rr * LST + half * 16;
#pragma unroll
        for (int i = 0; i < 4; ++i) {
            *(f4*)(la + i * 4) = *(const f4*)(pa + i * 4);
            *(f4*)(lb + i * 4) = *(const f4*)(pb + i * 4);
        }
        __syncthreads();

#pragma unroll
        for (int ks = 0; ks < 8; ++ks) {
            const int kk = ks * 4 + kh * 2;
            v2f a0 = *(const v2f*)(lA + (wr * 32 +      l15) * LST + kk);
            v2f a1 = *(const v2f*)(lA + (wr * 32 + 16 + l15) * LST + kk);
            v2f b[4];
#pragma unroll
            for (int j = 0; j < 4; ++j)
                b[j] = *(const v2f*)(lB + (wc * 64 + j * 16 + l15) * LST + kk);
#pragma unroll
            for (int j = 0; j < 4; ++j) {
                acc[0][j] = __builtin_amdgcn_wmma_f32_16x16x4_f32(
                    false, a0, false, b[j], (short)0, acc[0][j], false, false);
                acc[1][j] = __builtin_amdgcn_wmma_f32_16x16x4_f32(
                    false, a1, false, b[j], (short)0, acc[1][j], false, false);
            }
        }
        __syncthreads();
    }

    const float scale = ((by < 32) != (bx < 32)) ? 0.5f : 1.0f;

#pragma unroll
    for (int i = 0; i < 2; ++i)
#pragma unroll
        for (int j = 0; j < 4; ++j) {
            const int gr = rowBase + wr * 32 + i * 16 + 8 * kh;  // + v below
            const int gc = colBase + wc * 64 + j * 16 + l15;
            float* p = S + (size_t)gr * NN + gc;
#pragma unroll
            for (int v = 0; v < 8; ++v)
                p[(size_t)v * NN] = acc[i][j][v] * scale;
        }
}

// ---------------------------------------------------------------------------
// K3: per-row top-16 (diag forced), write sparsified raw row in place,
// record (idx,val) pairs and deg^-1/2. One block (256 threads = 8 waves)
// per row. Selection rounds use wave32 shfl_xor reduction (no barriers)
// + tiny 8-way cross-wave merge (3 barriers/round instead of ~8).
// ---------------------------------------------------------------------------
__device__ __forceinline__ void argmax2(float& bv, int& bi, float ov, int oi) {
    if (ov > bv || (ov == bv && oi < bi)) { bv = ov; bi = oi; }
}

__global__ __launch_bounds__(256) void topk_kernel(
    float* __restrict__ S, float* __restrict__ dinv,
    int* __restrict__ selIdx, float* __restrict__ selVal)
{
    const int row = blockIdx.x;
    const int t   = threadIdx.x;
    float* rp = S + (size_t)row * NN;

    float vals[32], orig[32];
#pragma unroll
    for (int j = 0; j < 32; ++j) {
        float v = rp[j * 256 + t];        // coalesced: col = j*256 + t
        orig[j] = v; vals[j] = v;
    }

    __shared__ float sv[8];
    __shared__ int   si[8];
    __shared__ float s_win;
    __shared__ int   s_wini;
    __shared__ float s_diag;
    __shared__ float s_selV[TOPK];
    __shared__ int   s_selI[TOPK];

    // exclude diagonal (always kept; ranked as +inf in reference)
    if ((row & 255) == t) {
        int j = row >> 8;
        s_diag = orig[j];
        vals[j] = SENTINEL;
    }
    __syncthreads();

    for (int r = 0; r < TOPK - 1; ++r) {
        float bv = SENTINEL; int bi = 0x7fffffff;
#pragma unroll
        for (int j = 0; j < 32; ++j) {
            int cidx = j * 256 + t;
            argmax2(bv, bi, vals[j], cidx);
        }
        // wave32 reduction, no barriers
#pragma unroll
        for (int off = 16; off >= 1; off >>= 1) {
            float ov = __shfl_xor(bv, off, 32);
            int   oi = __shfl_xor(bi, off, 32);
            argmax2(bv, bi, ov, oi);
        }
        if ((t & 31) == 0) { sv[t >> 5] = bv; si[t >> 5] = bi; }
        __syncthreads();
        if (t < 32) {                      // first wave merges the 8 leaders
            float mv = (t < 8) ? sv[t] : SENTINEL;
            int   mi = (t < 8) ? si[t] : 0x7fffffff;
#pragma unroll
            for (int off = 4; off >= 1; off >>= 1) {
                float ov = __shfl_xor(mv, off, 32);
                int   oi = __shfl_xor(mi, off, 32);
                argmax2(mv, mi, ov, oi);
            }
            if (t == 0) {
                s_win = mv; s_wini = mi;
                s_selI[r + 1] = mi; s_selV[r + 1] = mv;
            }
        }
        __syncthreads();
        const int wcol = s_wini;
        if ((wcol & 255) == t) vals[wcol >> 8] = SENTINEL;  // exclude winner
        __syncthreads();
    }

    if (t == 0) {
        s_selI[0] = row; s_selV[0] = s_diag;
        float deg = s_diag;
        for (int r = 1; r < TOPK; ++r) deg += s_selV[r];
        deg = fmaxf(deg, EPSV);
        dinv[row] = 1.0f / sqrtf(deg);
        for (int r = 0; r < TOPK; ++r) {
            selIdx[row * TOPK + r] = s_selI[r];
            selVal[row * TOPK + r] = s_selV[r];
        }
    }
    __syncthreads();

    // write sparsified (unnormalized) row: selected slots were sentinel-marked
#pragma unroll
    for (int j = 0; j < 32; ++j)
        rp[j * 256 + t] = (vals[j] == SENTINEL) ? orig[j] : 0.0f;
}

// ---------------------------------------------------------------------------
// K4: normalize kept entries in place: A[i,j] = v * dinv[i] * dinv[j]
// ---------------------------------------------------------------------------
__global__ __launch_bounds__(256) void scatter_kernel(
    float* __restrict__ S, const float* __restrict__ dinv,
    const int* __restrict__ selIdx, const float* __restrict__ selVal)
{
    const int idx = blockIdx.x * 256 + threadIdx.x;   // 8192*16 total
    const int row = idx >> 4;
    const int col = selIdx[idx];
    const float v = selVal[idx];
    S[(size_t)row * NN + col] = v * dinv[row] * dinv[col];
}

// ---------------------------------------------------------------------------
extern "C" void kernel_launch(void* const* d_in, const int* in_sizes, int n_in,
                              void* d_out, int out_size, void* d_ws, size_t ws_size,
                              hipStream_t stream) {
    const float* feats   = (const float*)d_in[0];   // [4096,256]
    const float* coords  = (const float*)d_in[1];   // [4096,4]
    const float* hfeats  = (const float*)d_in[2];   // [128,32,256]
    const float* hcoords = (const float*)d_in[3];   // [128,32,4]

    float* out   = (float*)d_out;
    float* adj   = out;                              // [8192*8192]
    float* nodes = out + (size_t)NN * NN;            // [8192*256]

    // workspace layout (~10.5 MB)
    float* G      = (float*)d_ws;                    // 8192*288 floats
    float* dinv   = G + (size_t)NN * CG;             // 8192 floats
    float* selVal = dinv + NN;                       // 8192*16 floats
    int*   selIdx = (int*)(selVal + (size_t)NN * TOPK);

    prep_kernel<<<NN, 256, 0, stream>>>(feats, coords, hfeats, hcoords, G, nodes);
    gemm_kernel<<<dim3(64, 64), 256, 0, stream>>>(G, adj);
    topk_kernel<<<NN, 256, 0, stream>>>(adj, dinv, selIdx, selVal);
    scatter_kernel<<<(NN * TOPK) / 256, 256, 0, stream>>>(adj, dinv, selIdx, selVal);
}